// EdgeEmbAttentionAggregator_18923625906529
// MI455X (gfx1250) — compile-verified
//
#include <hip/hip_runtime.h>
#include <hip/hip_bf16.h>
#include <math.h>

#define N_NODES   4096
#define IN_DIM    256
#define OUT_DIM   128
#define EDGE_DIM  64
#define N_SAMPLE  128
#define ALPHA     0.2f

typedef __attribute__((ext_vector_type(2))) float v2f;
typedef __attribute__((ext_vector_type(8))) float v8f;

// ---------------------------------------------------------------------------
// Kernel 1: X = features @ W   (4096x256 @ 256x128), fp32 WMMA 16x16x4.
// One wave = one 16x16 tile of X. 8 waves per block cover N=128 for one M-strip.
// ---------------------------------------------------------------------------
__global__ void __launch_bounds__(256)
gemm_xw_kernel(const float* __restrict__ F, const float* __restrict__ W,
               float* __restrict__ X) {
    const int wave  = threadIdx.x >> 5;        // 0..7 -> N tile
    const int lane  = threadIdx.x & 31;
    const int mBase = blockIdx.x * 16;         // 256 blocks
    const int nBase = wave * 16;
    const int row   = mBase + (lane & 15);     // A row for this lane
    const int col   = nBase + (lane & 15);     // B/D col for this lane
    const int kb    = (lane >> 4) * 2;         // which k-pair this half-wave holds

    v8f c = {};
    #pragma unroll 4
    for (int k = 0; k < IN_DIM; k += 4) {
        v2f a, b;
        a.x = F[row * IN_DIM + k + kb];
        a.y = F[row * IN_DIM + k + kb + 1];
        b.x = W[(k + kb)     * OUT_DIM + col];
        b.y = W[(k + kb + 1) * OUT_DIM + col];
        c = __builtin_amdgcn_wmma_f32_16x16x4_f32(false, a, false, b,
                                                  (short)0, c, false, false);
    }
    union { v8f v; float f[8]; } u; u.v = c;
    const int mHalf = (lane >> 4) * 8;
    #pragma unroll
    for (int v = 0; v < 8; ++v)
        X[(mBase + mHalf + v) * OUT_DIM + col] = u.f[v];
}

// ---------------------------------------------------------------------------
// Kernel 2: per-node attention. One 128-thread block per node.
//   e[s]   = leaky_relu(dot(x[n],a1) + dot(x[nb_s],a2) + dot(emb[eid_s],a3))
//   att    = softmax(e)
//   out[d] = elu(sum_s att[s]*x[nb_s][d] + x[n][d])
// All gathers hit L2 (x = 2MB, emb = 16MB resident).
// ---------------------------------------------------------------------------
__device__ __forceinline__ float block_reduce_sum(float v, float* red, int t) {
    red[t] = v; __syncthreads();
    #pragma unroll
    for (int off = 64; off > 0; off >>= 1) {
        if (t < off) red[t] += red[t + off];
        __syncthreads();
    }
    float r = red[0]; __syncthreads();
    return r;
}
__device__ __forceinline__ float block_reduce_max(float v, float* red, int t) {
    red[t] = v; __syncthreads();
    #pragma unroll
    for (int off = 64; off > 0; off >>= 1) {
        if (t < off) red[t] = fmaxf(red[t], red[t + off]);
        __syncthreads();
    }
    float r = red[0]; __syncthreads();
    return r;
}

__global__ void __launch_bounds__(128)
attention_kernel(const float* __restrict__ X, const int* __restrict__ e_index,
                 const int* __restrict__ neigh, const float* __restrict__ emb,
                 const float* __restrict__ a_vec, float* __restrict__ out) {
    __shared__ float a_lds[2 * OUT_DIM + EDGE_DIM];  // 320
    __shared__ float att[N_SAMPLE];
    __shared__ float red[N_SAMPLE];
    __shared__ int   nb_s[N_SAMPLE];

    const int n = blockIdx.x;
    const int t = threadIdx.x;

    for (int i = t; i < 2 * OUT_DIM + EDGE_DIM; i += 128) a_lds[i] = a_vec[i];
    const int nb = neigh[n * N_SAMPLE + t];
    nb_s[t] = nb;
    __syncthreads();

    // s1 = dot(x[n], a[0:128])  (shared across all samples of this node)
    const float xn_t = X[n * OUT_DIM + t];
    const float s1 = block_reduce_sum(xn_t * a_lds[t], red, t);

    // logits: thread t owns sample t
    float acc = s1;
    const float4* xr = (const float4*)(X + (size_t)nb * OUT_DIM);
    #pragma unroll 4
    for (int k = 0; k < OUT_DIM / 4; ++k) {
        float4 xv = xr[k];
        acc += xv.x * a_lds[OUT_DIM + 4 * k + 0];
        acc += xv.y * a_lds[OUT_DIM + 4 * k + 1];
        acc += xv.z * a_lds[OUT_DIM + 4 * k + 2];
        acc += xv.w * a_lds[OUT_DIM + 4 * k + 3];
    }
    const int eid = e_index[n * N_NODES + nb];
    const float4* er = (const float4*)(emb + (size_t)eid * EDGE_DIM);
    #pragma unroll 4
    for (int k = 0; k < EDGE_DIM / 4; ++k) {
        float4 ev = er[k];
        acc += ev.x * a_lds[2 * OUT_DIM + 4 * k + 0];
        acc += ev.y * a_lds[2 * OUT_DIM + 4 * k + 1];
        acc += ev.z * a_lds[2 * OUT_DIM + 4 * k + 2];
        acc += ev.w * a_lds[2 * OUT_DIM + 4 * k + 3];
    }
    float e = (acc > 0.0f) ? acc : ALPHA * acc;          // leaky_relu

    // softmax over 128 samples
    const float m  = block_reduce_max(e, red, t);
    const float ex = expf(e - m);
    const float sm = block_reduce_sum(ex, red, t);
    att[t] = ex / sm;
    __syncthreads();

    // aggregation: thread t owns output dim t (coalesced gather rows)
    float h = 0.0f;
    for (int s = 0; s < N_SAMPLE; ++s)
        h += att[s] * X[(size_t)nb_s[s] * OUT_DIM + t];
    float o = h + xn_t;
    o = (o > 0.0f) ? o : (expf(o) - 1.0f);               // elu
    out[n * OUT_DIM + t] = o;
}

// ---------------------------------------------------------------------------
// Kernel 3: edge embedding update, fused dual GEMM on WMMA f32.
//   D = (out[i]+out[j]) @ W2  +  emb[eid] @ W3        (16-edge tiles)
//   emb_new[k+1] = relu((D + B) * emb[k+1])
// 128 threads = 4 waves; wave w handles output cols [16w, 16w+16).
// ---------------------------------------------------------------------------
#define APAD 129   // 16x128 A tile, padded row stride
#define EPAD 65    // 16x64  E tile, padded row stride

__global__ void __launch_bounds__(128)
edge_kernel(const float* __restrict__ out, const float* __restrict__ emb,
            const int* __restrict__ e_index, const int* __restrict__ to_update,
            const float* __restrict__ W2, const float* __restrict__ W3,
            const float* __restrict__ Bvec, float* __restrict__ emb_new,
            int n_edges) {
    __shared__ float Asum[16 * APAD];
    __shared__ float Ae[16 * EPAD];

    const int t  = threadIdx.x;
    const int e0 = blockIdx.x * 16;

    // stage gathered rows: Asum[r] = out[i_r] + out[j_r], Ae[r] = emb[eid_r]
    for (int r = 0; r < 16; ++r) {
        const int k = e0 + r;
        const bool valid = (k < n_edges);
        int i = 0, j = 0, eid = 0;
        if (valid) {
            i   = to_update[2 * k];
            j   = to_update[2 * k + 1];
            eid = e_index[(size_t)i * N_NODES + j];
        }
        Asum[r * APAD + t] = valid ? (out[(size_t)i * OUT_DIM + t] +
                                      out[(size_t)j * OUT_DIM + t]) : 0.0f;
        if (t < EDGE_DIM)
            Ae[r * EPAD + t] = valid ? emb[(size_t)eid * EDGE_DIM + t] : 0.0f;
    }
    __syncthreads();

    const int wave = t >> 5;
    const int lane = t & 31;
    const int colN = wave * 16 + (lane & 15);
    const int rA   = lane & 15;
    const int kb   = (lane >> 4) * 2;

    v8f c = {};
    #pragma unroll 4
    for (int k = 0; k < OUT_DIM; k += 4) {        // v_input = Asum @ W2
        v2f a, b;
        a.x = Asum[rA * APAD + k + kb];
        a.y = Asum[rA * APAD + k + kb + 1];
        b.x = W2[(k + kb)     * EDGE_DIM + colN];
        b.y = W2[(k + kb + 1) * EDGE_DIM + colN];
        c = __builtin_amdgcn_wmma_f32_16x16x4_f32(false, a, false, b,
                                                  (short)0, c, false, false);
    }
    #pragma unroll 4
    for (int k = 0; k < EDGE_DIM; k += 4) {       // += e_input = Ae @ W3
        v2f a, b;
        a.x = Ae[rA * EPAD + k + kb];
        a.y = Ae[rA * EPAD + k + kb + 1];
        b.x = W3[(k + kb)     * EDGE_DIM + colN];
        b.y = W3[(k + kb + 1) * EDGE_DIM + colN];
        c = __builtin_amdgcn_wmma_f32_16x16x4_f32(false, a, false, b,
                                                  (short)0, c, false, false);
    }

    const float bcol  = Bvec[colN];
    const int   mHalf = (lane >> 4) * 8;
    union { v8f v; float f[8]; } u; u.v = c;
    #pragma unroll
    for (int v = 0; v < 8; ++v) {
        const int k = e0 + mHalf + v;
        if (k < n_edges) {
            const float val = u.f[v] + bcol;
            const float res = val * emb[(size_t)(k + 1) * EDGE_DIM + colN];
            emb_new[(size_t)(k + 1) * EDGE_DIM + colN] = res > 0.0f ? res : 0.0f;
        }
    }
}

// emb_new[0] = relu(B * emb[0])
__global__ void row0_kernel(const float* __restrict__ emb,
                            const float* __restrict__ Bvec,
                            float* __restrict__ emb_new) {
    const int t = threadIdx.x;
    const float r = Bvec[t] * emb[t];
    emb_new[t] = r > 0.0f ? r : 0.0f;
}

// ---------------------------------------------------------------------------
extern "C" void kernel_launch(void* const* d_in, const int* in_sizes, int n_in,
                              void* d_out, int out_size, void* d_ws, size_t ws_size,
                              hipStream_t stream) {
    const float* features = (const float*)d_in[0];  // 4096 x 256
    const int*   e_index  = (const int*)  d_in[1];  // 4096 x 4096
    const int*   neigh    = (const int*)  d_in[2];  // 4096 x 128
    const int*   to_upd   = (const int*)  d_in[3];  // n_edges x 2
    const float* emb      = (const float*)d_in[4];  // (n+1) x 64
    const float* W        = (const float*)d_in[5];  // 256 x 128
    const float* W2       = (const float*)d_in[6];  // 128 x 64
    const float* W3       = (const float*)d_in[7];  // 64 x 64
    const float* Bvec     = (const float*)d_in[8];  // 1 x 64
    const float* a_vec    = (const float*)d_in[9];  // 320 x 1

    float* out_feat = (float*)d_out;                       // 4096*128
    float* emb_new  = out_feat + (size_t)N_NODES * OUT_DIM;
    float* Xws      = (float*)d_ws;                        // scratch x = F@W

    const int n_edges = in_sizes[3] / 2;

    gemm_xw_kernel<<<N_NODES / 16, 256, 0, stream>>>(features, W, Xws);
    attention_kernel<<<N_NODES, 128, 0, stream>>>(Xws, e_index, neigh, emb,
                                                  a_vec, out_feat);
    edge_kernel<<<(n_edges + 15) / 16, 128, 0, stream>>>(out_feat, emb, e_index,
                                                         to_upd, W2, W3, Bvec,
                                                         emb_new, n_edges);
    row0_kernel<<<1, EDGE_DIM, 0, stream>>>(emb, Bvec, emb_new);
}